// ScaledDotProductAttention_34248069218851
// MI455X (gfx1250) — compile-verified
//
#include <hip/hip_runtime.h>
#include <hip/hip_bf16.h>
#include <math.h>
#include <stdint.h>

// ---------------------------------------------------------------------------
// Flash attention for B=4, H=16, S=2048, D=64, fp32, causal + key-padding mask.
// - Matrix math on V_WMMA_F32_16X16X4_F32 (native fp32 WMMA, wave32)
// - K/V tiles staged with GLOBAL_LOAD_ASYNC_TO_LDS (ASYNCcnt), double-buffered
// ---------------------------------------------------------------------------

#define B_  4
#define H_  16
#define S_  2048
#define D_  64

typedef float v2f __attribute__((ext_vector_type(2)));
typedef float v8f __attribute__((ext_vector_type(8)));

// 1/sqrt(64) * log2(e): scores computed directly in log2 domain
#define QSCALE (0.125f * 1.4426950408889634f)

// Tile/LDS geometry (floats). 32-key tiles, double buffered.
#define TK   32          // keys per tile
#define KST  68          // K tile row stride   [key][d],  64 + 4 pad
#define VST  36          // V tile row stride   [d][key],  32 + 4 pad
#define PST  20          // P scratch row stride, 16 + 4 pad

__device__ __forceinline__ float half16_max(float x) {
    x = fmaxf(x, __shfl_xor(x, 1, 32));
    x = fmaxf(x, __shfl_xor(x, 2, 32));
    x = fmaxf(x, __shfl_xor(x, 4, 32));
    x = fmaxf(x, __shfl_xor(x, 8, 32));
    return x;
}
__device__ __forceinline__ float half16_sum(float x) {
    x += __shfl_xor(x, 1, 32);
    x += __shfl_xor(x, 2, 32);
    x += __shfl_xor(x, 4, 32);
    x += __shfl_xor(x, 8, 32);
    return x;
}

__device__ __forceinline__ uint32_t lds_addr32(const void* p) {
    // generic address of a __shared__ object keeps the LDS byte offset in [31:0]
    return (uint32_t)(uintptr_t)p;
}

// --- kernel 0: per-(b,h) column mean of V (fallback for fully-masked rows) --
__global__ void fa_vmean_kernel(const float* __restrict__ v,
                                float* __restrict__ vmean) {
    const int bh = blockIdx.x;       // 0..63
    const int d  = threadIdx.x;      // 0..63
    const float* vp = v + (size_t)bh * S_ * D_ + d;
    float s = 0.0f;
    for (int i = 0; i < S_; ++i) s += vp[(size_t)i * D_];
    vmean[bh * D_ + d] = s * (1.0f / (float)S_);
}

// --- main flash-attention kernel -------------------------------------------
// grid = (S/64, B*H), block = 128 (4 waves); wave w owns 16 query rows.
__global__ __launch_bounds__(128)
void fa_fwd_kernel(const float* __restrict__ kmat,
                   const float* __restrict__ qmat,
                   const float* __restrict__ vmat,
                   const unsigned char* __restrict__ amask, // (B,S) bool
                   const int* __restrict__ causal_flag,     // scalar
                   const float* __restrict__ vmean,         // (B*H, D)
                   float* __restrict__ out) {
    __shared__ float Klds[2][TK * KST];       // K tile, row-major [key][d]
    __shared__ float Vt  [2][D_ * VST];       // V tile, transposed [d][key]
    __shared__ float Plds[4 * 16 * PST];      // per-wave P scratch

    const int tid  = threadIdx.x;
    const int wave = tid >> 5;
    const int lane = tid & 31;
    const int lm   = lane & 15;
    const int half = lane >> 4;

    const int bh = blockIdx.y;                // 0..63
    const int b  = bh >> 4;                   // /H_
    const int q0 = blockIdx.x * 64;
    const int qbase = q0 + wave * 16;

    const int causal = causal_flag[0];

    const unsigned char* amp = amask + (size_t)b * S_;
    const float* kgbase = kmat + (size_t)bh * S_ * D_;
    const float* vgbase = vmat + (size_t)bh * S_ * D_;
    float* Pw = &Plds[wave * 16 * PST];

    // ---- load Q tile (16 rows per wave) as pre-scaled WMMA A fragments ----
    // A layout (f32 16x16x4): lane l holds M=l%16, K = v + 2*(l/16)
    const int qrow = qbase + lm;
    const float* qp = qmat + ((size_t)bh * S_ + qrow) * D_;
    v2f qa[16];
#pragma unroll
    for (int c = 0; c < 16; ++c) {
        v2f t = *(const v2f*)(qp + 4 * c + 2 * half);
        qa[c] = t * QSCALE;
    }

    // ---- running softmax state (log2 domain) and O accumulators ----
    float m_i[8], l_i[8];
#pragma unroll
    for (int r = 0; r < 8; ++r) { m_i[r] = -INFINITY; l_i[r] = 0.0f; }
    v8f acc[4];
#pragma unroll
    for (int t = 0; t < 4; ++t) acc[t] = (v8f){};

    const int kend   = causal ? (q0 + 64) : S_;   // WG-uniform
    const int nTiles = kend / TK;

    // ---- async tile issue: K rows by b128, V transposed by per-lane b32 ----
    auto issue_tile = [&](int tile, int buf) {
        const float* kgp = kgbase + (size_t)tile * TK * D_;
        const float* vgp = vgbase + (size_t)tile * TK * D_;
#pragma unroll
        for (int i = tid; i < TK * 16; i += 128) {        // 512 float4s of K
            const int dv  = i & 15;                        // float4 column
            const int key = i >> 4;                        // 0..31
            const uint32_t la = lds_addr32(&Klds[buf][key * KST + dv * 4]);
            const uint32_t go = (uint32_t)((key * D_ + dv * 4) * 4);
            asm volatile("global_load_async_to_lds_b128 %0, %1, %2"
                         :: "v"(la), "v"(go), "s"(kgp) : "memory");
        }
#pragma unroll
        for (int i = tid; i < TK * D_; i += 128) {        // 2048 floats of V
            const int d   = i & 63;                        // coalesced in d
            const int key = i >> 6;                        // 0..31
            const uint32_t la = lds_addr32(&Vt[buf][d * VST + key]);
            const uint32_t go = (uint32_t)((key * D_ + d) * 4);
            asm volatile("global_load_async_to_lds_b32 %0, %1, %2"
                         :: "v"(la), "v"(go), "s"(vgp) : "memory");
        }
    };

    issue_tile(0, 0);   // prologue

    for (int it = 0; it < nTiles; ++it) {
        const int buf = it & 1;
        const int n0  = it * TK;

        asm volatile("s_wait_asynccnt 0x0" ::: "memory");  // my tile loads done
        __syncthreads();                                   // everyone's done
        if (it + 1 < nTiles) issue_tile(it + 1, buf ^ 1);  // overlap next DMA

#pragma unroll
        for (int s16 = 0; s16 < 2; ++s16) {
            const int n0s = n0 + 16 * s16;
            if (causal && n0s > qbase + 15) break;  // wave-uniform skip
            const int kcol = n0s + lm;
            const unsigned char mb = amp[kcol];

            // ---- scores: 16x16 tile, 16 chained K=4 fp32 WMMAs ----
            v8f sc = (v8f){};
#pragma unroll
            for (int c = 0; c < 16; ++c) {
                v2f kb = *(const v2f*)(&Klds[buf][(s16 * 16 + lm) * KST +
                                                  4 * c + 2 * half]);
                sc = __builtin_amdgcn_wmma_f32_16x16x4_f32(
                        false, qa[c], false, kb, (short)0, sc, false, false);
            }

            // ---- online softmax update (log2 domain) ----
            float pr[8];
#pragma unroll
            for (int r = 0; r < 8; ++r) {
                const int mrow = qbase + r + 8 * half;
                const bool ok = (mb != 0) && (!causal || (kcol <= mrow));
                const float sv = sc[r];
                const float tmax = half16_max(ok ? sv : -INFINITY);
                const float mn = fmaxf(m_i[r], tmax);
                const float alpha =
                    (mn == -INFINITY) ? 1.0f : __builtin_amdgcn_exp2f(m_i[r] - mn);
                const float p = ok ? __builtin_amdgcn_exp2f(sv - mn) : 0.0f;
                const float rs = half16_sum(p);
                l_i[r] = l_i[r] * alpha + rs;
                m_i[r] = mn;
                pr[r]  = p;
#pragma unroll
                for (int t = 0; t < 4; ++t) acc[t][r] *= alpha;
            }

            // ---- re-layout P (C-layout -> A-layout) through per-wave LDS ----
#pragma unroll
            for (int r = 0; r < 8; ++r)
                Pw[(r + 8 * half) * PST + lm] = pr[r];
            asm volatile("s_wait_dscnt 0x0" ::: "memory");  // wave-local fence

            // ---- O += P * V : 16 chained K=4 fp32 WMMAs ----
#pragma unroll
            for (int cc = 0; cc < 4; ++cc) {
                v2f pa = *(const v2f*)(&Pw[lm * PST + 4 * cc + 2 * half]);
#pragma unroll
                for (int t = 0; t < 4; ++t) {
                    v2f vb = *(const v2f*)(&Vt[buf][(16 * t + lm) * VST +
                                                    s16 * 16 + 4 * cc + 2 * half]);
                    acc[t] = __builtin_amdgcn_wmma_f32_16x16x4_f32(
                                false, pa, false, vb, (short)0, acc[t], false, false);
                }
            }
        }
    }

    // ---- epilogue: normalize; fully-masked rows -> mean over all of V ----
    float vmcol[4];
#pragma unroll
    for (int t = 0; t < 4; ++t) vmcol[t] = vmean[bh * D_ + 16 * t + lm];
#pragma unroll
    for (int r = 0; r < 8; ++r) {
        const int row = qbase + r + 8 * half;
        const float li = l_i[r];
        const bool dead = !(li > 0.0f);
        const float inv = dead ? 0.0f : 1.0f / li;
        float* op = out + ((size_t)bh * S_ + row) * D_;
#pragma unroll
        for (int t = 0; t < 4; ++t) {
            const float val = dead ? vmcol[t] : acc[t][r] * inv;
            op[16 * t + lm] = val;
        }
    }
}

// ---------------------------------------------------------------------------
extern "C" void kernel_launch(void* const* d_in, const int* in_sizes, int n_in,
                              void* d_out, int out_size, void* d_ws, size_t ws_size,
                              hipStream_t stream) {
    const float* k = (const float*)d_in[0];
    const float* q = (const float*)d_in[1];
    const float* v = (const float*)d_in[2];
    const unsigned char* am = (const unsigned char*)d_in[3];  // jax bool = 1 byte
    const int* causal = (const int*)d_in[4];
    float* out = (float*)d_out;
    float* vmean = (float*)d_ws;   // B*H*D floats = 16 KB

    fa_vmean_kernel<<<dim3(B_ * H_), dim3(D_), 0, stream>>>(v, vmean);

    dim3 grid(S_ / 64, B_ * H_);
    dim3 block(128);
    fa_fwd_kernel<<<grid, block, 0, stream>>>(k, q, v, am, causal, vmean, out);
}